// QGCost_new_60945585930615
// MI455X (gfx1250) — compile-verified
//
#include <hip/hip_runtime.h>

#define GG   9.81f
#define DTT  200.0f
#define KDD  100.0f
#define NPX  400          // physical grid
#define NI   398          // interior / DST size
#define NB   8            // batch
#define NPAD 448          // padded DST workspace (28 tiles of 16, 7 blocks of 64)
#define KC   32           // K chunk staged per double-buffer step
#define NCH  (NPAD / KC)  // 14 chunks
#define ASTR 36           // LDS stride for A tile (36*2 % 64 == 8 -> conflict-free b64)
#define BSTR 80           // LDS stride for B tile (80*2 % 64 == 32 -> conflict-free b32)
#define PI_D 3.14159265358979323846

typedef float v2f __attribute__((ext_vector_type(2)));
typedef float v8f __attribute__((ext_vector_type(8)));

static __device__ __forceinline__ v8f wmma4(v2f a, v2f b, v8f c) {
  return __builtin_amdgcn_wmma_f32_16x16x4_f32(false, a, false, b, (short)0, c,
                                               false, false);
}

// 16B global -> LDS async copy (ASYNCcnt-tracked, CDNA5 ISA 15.18.3 op 98)
static __device__ __forceinline__ void async_copy16(const float* g, float* l) {
  __attribute__((address_space(3))) float* lp =
      (__attribute__((address_space(3))) float*)l;
  asm volatile("global_load_async_to_lds_b128 %0, %1, off" ::"v"(lp), "v"(g)
               : "memory");
}

// ---------------------------------------------------------------- DST matrix
__global__ __launch_bounds__(256) void build_S(float* __restrict__ S) {
  int idx = blockIdx.x * blockDim.x + threadIdx.x;
  if (idx >= NPAD * NPAD) return;
  int j = idx / NPAD, k = idx % NPAD;
  float v = 0.f;
  if (j < NI && k < NI) {
    long m = ((long)(j + 1) * (long)(k + 1)) % (2 * (NI + 1));
    double ang = PI_D * (double)m / (double)(NI + 1);
    v = (float)(sqrt(2.0 / (double)(NI + 1)) * sin(ang));
  }
  S[idx] = v;
}

// ------------------------------------------------------- initial state: q,qb,h
__global__ __launch_bounds__(256)
void init_state(const float* __restrict__ h0, const float* __restrict__ f,
                const float* __restrict__ c, const float* __restrict__ dxp,
                const unsigned char* __restrict__ ind0,
                const unsigned char* __restrict__ ind12,
                float* __restrict__ h, float* __restrict__ q,
                float* __restrict__ qb) {
  int idx = blockIdx.x * blockDim.x + threadIdx.x;
  if (idx >= NPX * NPX) return;
  int b = blockIdx.y;
  int i = idx / NPX, j = idx % NPX;
  size_t off = (size_t)b * NPX * NPX;
  const float* hb = h0 + off;
  float dx = dxp[0];
  float dx2 = dx * dx;
  float hv = hb[idx];
  h[off + idx] = hv;
  float qv = 0.f;
  if (i >= 1 && i <= NPX - 2 && j >= 1 && j <= NPX - 2) {
    float lap = (hb[idx + NPX] + hb[idx - NPX] - 2.f * hv) / dx2 +
                (hb[idx + 1] + hb[idx - 1] - 2.f * hv) / dx2;
    qv = GG / f[idx] * lap - GG * f[idx] / (c[idx] * c[idx]) * hv;
    if (qv != qv) qv = 0.f;
  }
  if (ind12[idx]) qv = -GG * f[idx] / (c[idx] * c[idx]) * hv;
  if (ind0[idx]) qv = 0.f;
  q[off + idx] = qv;
  qb[off + idx] = qv;
}

// ------------------------------------------------------------------- h2uv
__global__ __launch_bounds__(256)
void h2uv(const float* __restrict__ h, const float* __restrict__ f,
          const float* __restrict__ dxp, float* __restrict__ u,
          float* __restrict__ v) {
  int idx = blockIdx.x * blockDim.x + threadIdx.x;
  if (idx >= NPX * NPX) return;
  int b = blockIdx.y;
  int i = idx / NPX, j = idx % NPX;
  size_t off = (size_t)b * NPX * NPX;
  const float* hh = h + off;
  float inv4 = 1.f / (4.f * dxp[0]);
  float uv = 0.f, vv = 0.f;
  if (i >= 1 && i <= NPX - 2 && j >= 1) {
    uv = -GG / f[idx] *
         (hh[idx + NPX - 1] + hh[idx + NPX] - hh[idx - NPX] - hh[idx - NPX - 1]) *
         inv4;
    if (uv != uv) uv = 0.f;
  }
  if (i >= 1 && j >= 1 && j <= NPX - 2) {
    vv = GG / f[idx] *
         (hh[idx + 1] + hh[idx - NPX + 1] - hh[idx - NPX - 1] - hh[idx - 1]) *
         inv4;
    if (vv != vv) vv = 0.f;
  }
  u[off + idx] = uv;
  v[off + idx] = vv;
}

// ----------------------------------------------- rhs (adv+beta+diff) + q step
__global__ __launch_bounds__(256)
void rhs_qstep(const float* __restrict__ qold, const float* __restrict__ u,
               const float* __restrict__ v, const float* __restrict__ f,
               const float* __restrict__ dxp,
               const unsigned char* __restrict__ ind0,
               const unsigned char* __restrict__ ind12,
               float* __restrict__ qnew) {
  int idx = blockIdx.x * blockDim.x + threadIdx.x;
  if (idx >= NPX * NPX) return;
  int b = blockIdx.y;
  int i = idx / NPX, j = idx % NPX;
  size_t off = (size_t)b * NPX * NPX;
  const float* q = qold + off;
  const float* uu = u + off;
  const float* vv = v + off;
  float dx = dxp[0];
  float r = 0.f;
  if (i >= 2 && i <= NPX - 3 && j >= 2 && j <= NPX - 3) {
    float uT = 0.5f * (uu[idx] + uu[idx + 1]);
    float vT = 0.5f * (vv[idx] + vv[idx + NPX]);
    float up = uT < 0.f ? 0.f : uT;
    float um = uT > 0.f ? 0.f : uT;
    float vp = vT < 0.f ? 0.f : vT;
    float vm = vT > 0.f ? 0.f : vT;
    float i6 = 1.f / (6.f * dx);
    float adv =
        -up * i6 * (2.f * q[idx + 1] + 3.f * q[idx] - 6.f * q[idx - 1] + q[idx - 2]) +
        um * i6 * (q[idx + 2] - 6.f * q[idx + 1] + 3.f * q[idx] + 2.f * q[idx - 1]) -
        vp * i6 * (2.f * q[idx + NPX] + 3.f * q[idx] - 6.f * q[idx - NPX] +
                   q[idx - 2 * NPX]) +
        vm * i6 * (q[idx + 2 * NPX] - 6.f * q[idx + NPX] + 3.f * q[idx] +
                   2.f * q[idx - NPX]);
    float beta = -(f[idx + NPX] - f[idx - NPX]) / (2.f * dx) * 0.5f *
                 (vv[idx] + vv[idx + NPX]);
    float kd2 = KDD / (dx * dx);
    float diff = kd2 * (q[idx + 1] + q[idx - 1] - 2.f * q[idx]) +
                 kd2 * (q[idx + NPX] + q[idx - NPX] - 2.f * q[idx]);
    r = adv + beta + diff;
    if (r != r) r = 0.f;
  }
  if (ind12[idx] | ind0[idx]) r = 0.f;
  qnew[off + idx] = q[idx] + DTT * r; // WAY = 1
}

// ------------------------------- qin = (q - qb) interior, zero-padded to 448^2
__global__ __launch_bounds__(256)
void make_qin(const float* __restrict__ q, const float* __restrict__ qb,
              float* __restrict__ qin) {
  int idx = blockIdx.x * blockDim.x + threadIdx.x;
  if (idx >= NPAD * NPAD) return;
  int b = blockIdx.y;
  int i = idx / NPAD, j = idx % NPAD;
  float vq = 0.f;
  if (i < NI && j < NI) {
    size_t s = (size_t)b * NPX * NPX + (i + 1) * NPX + (j + 1);
    vq = q[s] - qb[s];
  }
  qin[(size_t)b * NPAD * NPAD + idx] = vq;
}

// ---- cooperative async staging of one K-chunk: A 64xKC, B KCx64 -> LDS ------
static __device__ __forceinline__ void stage_tiles(const float* Ab,
                                                   const float* Bb, int m0,
                                                   int n0, int kc0, float* Asb,
                                                   float* Bsb, int t) {
#pragma unroll
  for (int it = 0; it < 4; ++it) {
    int li = it * 128 + t;
    int ar = li >> 3, ac = (li & 7) << 2;    // A: 64 rows x 32 k
    int br = li >> 4, bc = (li & 15) << 2;   // B: 32 k    x 64 cols
    async_copy16(Ab + (size_t)(m0 + ar) * NPAD + kc0 + ac, Asb + ar * ASTR + ac);
    async_copy16(Bb + (size_t)(kc0 + br) * NPAD + n0 + bc, Bsb + br * BSTR + bc);
  }
}

// --------------------------------------------------------------- WMMA f32 GEMM
// C_b = A_b * B_b on 448x448 zero-padded matrices. 128 threads / 4 waves per
// block; block tile 64x64, wave tile 32x32 (4 f32 accumulators). K staged in
// double-buffered LDS via global_load_async_to_lds_b128 + s_wait_asynccnt.
//   mode 0: C = A*B            (448-stride)
//   mode 1: C = (A*B)/helm     (448-stride, Helmholtz division in DST space)
//   mode 2: h[1+r][1+c] = h0[1+r][1+c] + (A*B)[r][c]   (400-stride finalize)
__global__ __launch_bounds__(128)
void dst_gemm(const float* __restrict__ A, long sAb,
              const float* __restrict__ B, long sBb, float* __restrict__ C,
              int mode, const float* __restrict__ helm,
              const float* __restrict__ hb) {
  __shared__ float As[2][64 * ASTR];
  __shared__ float Bs[2][KC * BSTR];

  int t = threadIdx.x;
  int lane = t & 31;
  int w = t >> 5;            // wave 0..3
  int wm = w >> 1, wn = w & 1;
  int half = lane >> 4, lr = lane & 15;

  int b = blockIdx.x / 49;
  int tile = blockIdx.x % 49;
  int m0 = (tile / 7) * 64;
  int n0 = (tile % 7) * 64;

  const float* Ab = A + (size_t)b * (size_t)sAb;
  const float* Bb = B + (size_t)b * (size_t)sBb;

  v8f acc[2][2];
#pragma unroll
  for (int ti = 0; ti < 2; ++ti)
#pragma unroll
    for (int tj = 0; tj < 2; ++tj) acc[ti][tj] = (v8f){0.f, 0.f, 0.f, 0.f,
                                                       0.f, 0.f, 0.f, 0.f};

  stage_tiles(Ab, Bb, m0, n0, 0, As[0], Bs[0], t);

  for (int c = 0; c < NCH; ++c) {
    if (c + 1 < NCH) {
      stage_tiles(Ab, Bb, m0, n0, (c + 1) * KC, As[(c + 1) & 1], Bs[(c + 1) & 1], t);
      asm volatile("s_wait_asynccnt 0x8" ::: "memory"); // chunk c retired (FIFO)
    } else {
      asm volatile("s_wait_asynccnt 0x0" ::: "memory");
    }
    __syncthreads();

    const float* Ac = As[c & 1];
    const float* Bc = Bs[c & 1];
    const float* ap = Ac + (wm * 32 + lr) * ASTR + 2 * half;
    const float* bp = Bc + (2 * half) * BSTR + wn * 32 + lr;
#pragma unroll 4
    for (int kk = 0; kk < KC; kk += 4) {
      v2f a0 = *(const v2f*)(ap + kk);
      v2f a1 = *(const v2f*)(ap + 16 * ASTR + kk);
      v2f b0, b1;
      b0.x = bp[kk * BSTR];
      b0.y = bp[kk * BSTR + BSTR];
      b1.x = bp[kk * BSTR + 16];
      b1.y = bp[kk * BSTR + BSTR + 16];
      acc[0][0] = wmma4(a0, b0, acc[0][0]);
      acc[0][1] = wmma4(a0, b1, acc[0][1]);
      acc[1][0] = wmma4(a1, b0, acc[1][0]);
      acc[1][1] = wmma4(a1, b1, acc[1][1]);
    }
    __syncthreads();
  }

  size_t coffP = (size_t)b * NPAD * NPAD;
  size_t coffH = (size_t)b * NPX * NPX;
#pragma unroll
  for (int ti = 0; ti < 2; ++ti)
#pragma unroll
    for (int tj = 0; tj < 2; ++tj)
#pragma unroll
      for (int g = 0; g < 8; ++g) {
        int r = m0 + wm * 32 + ti * 16 + g + 8 * half; // ISA C/D layout
        int cc = n0 + wn * 32 + tj * 16 + lr;
        float val = acc[ti][tj][g];
        if (mode == 1) {
          if (r < NI && cc < NI) val = val / helm[r * NI + cc];
          C[coffP + (size_t)r * NPAD + cc] = val;
        } else if (mode == 2) {
          if (r < NI && cc < NI)
            C[coffH + (size_t)(r + 1) * NPX + (cc + 1)] =
                hb[coffH + (size_t)(r + 1) * NPX + (cc + 1)] + val;
        } else {
          C[coffP + (size_t)r * NPAD + cc] = val;
        }
      }
}

// ----------------------------------------------------------------- launcher
extern "C" void kernel_launch(void* const* d_in, const int* in_sizes, int n_in,
                              void* d_out, int out_size, void* d_ws,
                              size_t ws_size, hipStream_t stream) {
  const float* h0 = (const float*)d_in[0];
  const float* f = (const float*)d_in[1];
  const float* c = (const float*)d_in[2];
  const float* helm = (const float*)d_in[3];
  const float* dxp = (const float*)d_in[4];
  const unsigned char* ind0 = (const unsigned char*)d_in[5];
  const unsigned char* ind12 = (const unsigned char*)d_in[6];
  float* h = (float*)d_out;

  const size_t NPP = (size_t)NPX * NPX;    // 160000
  const size_t BNP = NPP * NB;             // 1,280,000
  const size_t NPP2 = (size_t)NPAD * NPAD; // 200704
  const size_t BNP2 = NPP2 * NB;
  float* ws = (float*)d_ws;
  float* S = ws;   ws += NPP2;
  float* qa = ws;  ws += BNP;
  float* qb = ws;  ws += BNP;
  float* uu = ws;  ws += BNP;
  float* vv = ws;  ws += BNP;
  float* qc = ws;  ws += BNP; // ping-pong partner for qa
  float* qin = ws; ws += BNP2;
  float* T1 = ws;  ws += BNP2;
  float* T2 = ws;  ws += BNP2;

  dim3 blk(256);
  dim3 grd((unsigned)((NPP + 255) / 256), NB);
  dim3 grdP((unsigned)((NPP2 + 255) / 256), NB);
  dim3 grd1((unsigned)((NPP2 + 255) / 256));
  dim3 gblk(128);
  dim3 ggrd((unsigned)(NB * 49)); // 7x7 tiles of 64 per batch

  build_S<<<grd1, blk, 0, stream>>>(S);
  init_state<<<grd, blk, 0, stream>>>(h0, f, c, dxp, ind0, ind12, h, qa, qb);

  float* qcur = qa;
  float* qnxt = qc;
  for (int step = 0; step < 10; ++step) {
    h2uv<<<grd, blk, 0, stream>>>(h, f, dxp, uu, vv);
    rhs_qstep<<<grd, blk, 0, stream>>>(qcur, uu, vv, f, dxp, ind0, ind12, qnxt);
    make_qin<<<grdP, blk, 0, stream>>>(qnxt, qb, qin);
    // pv2h: inv = S * ((S*qin*S)/helm) * S ; h = hb + inv (interior)
    dst_gemm<<<ggrd, gblk, 0, stream>>>(qin, (long)NPP2, S, 0L, T1, 0, helm, h0);
    dst_gemm<<<ggrd, gblk, 0, stream>>>(S, 0L, T1, (long)NPP2, T2, 1, helm, h0);
    dst_gemm<<<ggrd, gblk, 0, stream>>>(T2, (long)NPP2, S, 0L, T1, 0, helm, h0);
    dst_gemm<<<ggrd, gblk, 0, stream>>>(S, 0L, T1, (long)NPP2, h, 2, helm, h0);
    float* tmp = qcur;
    qcur = qnxt;
    qnxt = tmp;
  }
}